// ETracking_Net_86526411145636
// MI455X (gfx1250) — compile-verified
//
// ETracking_Net for MI455X (gfx1250), fp32 end-to-end.
//
// Design notes (MI455X reasoning):
//  * Total work ~20 GFLOP / ~45 MB touched -> ~2us of HBM time at 23.3 TB/s.
//    The net is launch/latency bound, so precision stays fp32 (matches the
//    reference dtype) and all GEMM-shaped ops use V_WMMA_F32_16X16X4_F32.
//  * One batched WMMA GEMM kernel (64x64 tile, 128 threads = 4 wave32 waves,
//    each wave a 2x2 grid of 16x16 WMMA tiles, BK=16 staged in LDS K-major,
//    transA/transB resolved at LDS fill) services: fe-layer convs, attention
//    MLP, QKV convs, per-head Gram (K^T V, transA), and output (Q x H^T,
//    transB). Batch strides let z = b*8+h address per-head blocks directly.
//  * Tile staging uses GLOBAL_LOAD_ASYNC_TO_LDS_B32 (ASYNCcnt-tracked,
//    no VGPR round trip): builtin confirmed present; signature takes
//    addrspace(1)/addrspace(3) int* params (per round-2 diagnostic).
//    Explicit s_wait_asynccnt 0 drains before the workgroup barrier.
//  * kNN: one center per wave32, per-lane register top-K (unrolled insertion
//    chains, template K in {12,4}), 5-round __shfl_xor butterfly merge.
//  * jax threefry parity is not reproduced (compile-only loop; no runtime
//    check possible). Centers come from a deterministic injective map
//    s -> (s*1000003 + hash(b,seed)) mod P. The reference's torch
//    flatten/view index scramble idx[b, j%S, j/S] IS replicated.
//  * Workspace: bump-allocated from d_ws, ~73 MB fp32 (transformer buffers
//    alias extraction scratch). No hipMalloc / sync APIs (graph-capture safe).

#include <hip/hip_runtime.h>
#include <hip/hip_bf16.h>
#include <math.h>

typedef float v2f __attribute__((ext_vector_type(2)));
typedef float v8f __attribute__((ext_vector_type(8)));

#if defined(__AMDGCN__) &&                                                     \
    __has_builtin(__builtin_amdgcn_global_load_async_to_lds_b32) &&            \
    __has_builtin(__builtin_amdgcn_s_wait_asynccnt)
#define HAVE_ASYNC_LDS 1
typedef __attribute__((address_space(1))) int GLOBi;   // global (AS1) int*
typedef __attribute__((address_space(3))) int LDSi;    // LDS (AS3) int*
#else
#define HAVE_ASYNC_LDS 0
#endif

#define GEMM_BM 64
#define GEMM_BN 64
#define GEMM_BK 16

// ---------------------------------------------------------------------------
// Batched GEMM: C[z] = alpha * opA(A[z]) * opB(B[z])
//   opA(A)(m,k) = transA ? A[k*lda+m] : A[m*lda+k]
//   opB(B)(k,n) = transB ? B[n*ldb+k] : B[k*ldb+n]
// fp32 WMMA 16x16x4. 128 threads (4 waves), each wave owns a 32x32 subtile.
// ---------------------------------------------------------------------------
__global__ __launch_bounds__(128)
void k_gemm_f32(const float* __restrict__ A, const float* __restrict__ B,
                float* __restrict__ C,
                int M, int N, int K, int lda, int ldb, int ldc,
                long long sA, long long sB, long long sC,
                int transA, int transB, float alpha)
{
  __shared__ float As[GEMM_BK][GEMM_BM + 1];   // K-major: As[k][m]
  __shared__ float Bs[GEMM_BK][GEMM_BN + 1];   // K-major: Bs[k][n]
  const int z = blockIdx.z;
  A += (long long)z * sA; B += (long long)z * sB; C += (long long)z * sC;
  const int m_blk = blockIdx.y * GEMM_BM;
  const int n_blk = blockIdx.x * GEMM_BN;
  const int tid  = threadIdx.x;
  const int wave = tid >> 5, lane = tid & 31;
  const int wm = (wave >> 1) * 32, wn = (wave & 1) * 32;
  const int l16 = lane & 15, half = lane >> 4;

  v8f acc[2][2] = {};

  for (int k0 = 0; k0 < K; k0 += GEMM_BK) {
    // stage A tile (zero-padded at edges)
    for (int i = tid; i < GEMM_BM * GEMM_BK; i += 128) {
      int m = i / GEMM_BK, kk = i % GEMM_BK;
      int gm = m_blk + m, gk = k0 + kk;
      float* dst = &As[kk][m];
      if (gm < M && gk < K) {
        const float* src =
            transA ? &A[(long long)gk * lda + gm] : &A[(long long)gm * lda + gk];
#if HAVE_ASYNC_LDS
        __builtin_amdgcn_global_load_async_to_lds_b32(
            (GLOBi*)const_cast<float*>(src), (LDSi*)dst, 0, 0);
#else
        *dst = *src;
#endif
      } else {
        *dst = 0.f;
      }
    }
    // stage B tile
    for (int i = tid; i < GEMM_BN * GEMM_BK; i += 128) {
      int n = i % GEMM_BN, kk = i / GEMM_BN;
      int gn = n_blk + n, gk = k0 + kk;
      float* dst = &Bs[kk][n];
      if (gn < N && gk < K) {
        const float* src =
            transB ? &B[(long long)gn * ldb + gk] : &B[(long long)gk * ldb + gn];
#if HAVE_ASYNC_LDS
        __builtin_amdgcn_global_load_async_to_lds_b32(
            (GLOBi*)const_cast<float*>(src), (LDSi*)dst, 0, 0);
#else
        *dst = *src;
#endif
      } else {
        *dst = 0.f;
      }
    }
#if HAVE_ASYNC_LDS
    __builtin_amdgcn_s_wait_asynccnt(0);   // drain ASYNCcnt before barrier
#endif
    __syncthreads();

    // prefetch next K-slab (emits global_prefetch_b8)
    if (k0 + GEMM_BK < K) {
      long long nk = k0 + GEMM_BK;
      const float* pa = transA ? (A + nk * lda + m_blk)
                               : (A + (long long)m_blk * lda + nk);
      const float* pb = transB ? (B + (long long)n_blk * ldb + nk)
                               : (B + nk * ldb + n_blk);
      __builtin_prefetch(pa + tid, 0, 1);
      __builtin_prefetch(pb + tid, 0, 1);
    }

#pragma unroll
    for (int kk = 0; kk < GEMM_BK; kk += 4) {
      // A frag (16x4): lanes 0-15 carry K={kk,kk+1}, lanes 16-31 K={kk+2,kk+3}
      v2f af[2], bf[2];
#pragma unroll
      for (int i = 0; i < 2; ++i) {
        af[i].x = As[kk + 2 * half + 0][wm + i * 16 + l16];
        af[i].y = As[kk + 2 * half + 1][wm + i * 16 + l16];
      }
#pragma unroll
      for (int j = 0; j < 2; ++j) {
        bf[j].x = Bs[kk + 2 * half + 0][wn + j * 16 + l16];
        bf[j].y = Bs[kk + 2 * half + 1][wn + j * 16 + l16];
      }
#pragma unroll
      for (int i = 0; i < 2; ++i)
#pragma unroll
        for (int j = 0; j < 2; ++j)
          acc[i][j] = __builtin_amdgcn_wmma_f32_16x16x4_f32(
              false, af[i], false, bf[j], (short)0, acc[i][j], false, false);
    }
    __syncthreads();
  }

  // C/D layout: VGPR v -> M = v + 8*half (lanes 0-15: M=v, 16-31: M=v+8), N=l16
#pragma unroll
  for (int i = 0; i < 2; ++i) {
#pragma unroll
    for (int j = 0; j < 2; ++j) {
      int col = n_blk + wn + j * 16 + l16;
      if (col < N) {
#pragma unroll
        for (int v = 0; v < 8; ++v) {
          int row = m_blk + wm + i * 16 + half * 8 + v;
          if (row < M) C[(long long)row * ldc + col] = alpha * acc[i][j][v];
        }
      }
    }
  }
}

// ---------------------------------------------------------------------------
// Deterministic injective center sampling: perm[b][s] = (s*1000003 + off) % P
// ---------------------------------------------------------------------------
__global__ void k_perm(int* __restrict__ perm, int B, int P, int S, unsigned seed)
{
  int i = blockIdx.x * blockDim.x + threadIdx.x;
  if (i >= B * S) return;
  int b = i / S, s = i % S;
  unsigned off = (seed * 2654435761u + (unsigned)b * 40503u) % (unsigned)P;
  long long v = ((long long)s * 1000003ll) % P;
  perm[i] = (int)((v + off) % P);
}

// ---------------------------------------------------------------------------
// kNN: one center per wave32. pd = 2*inner - |q|^2 - |qs|^2 (maximize).
// Per-lane register top-K with unrolled insertion chains, then a 5-round
// __shfl_xor butterfly merge -> every lane holds the global top-K.
// ---------------------------------------------------------------------------
template <int KK_>
__global__ __launch_bounds__(256)
void k_knn(const float* __restrict__ x, const int* __restrict__ perm,
           int* __restrict__ knn, int B, int C, int Cq, int P, int S)
{
  __shared__ float csh[8][520];
  const int wave = threadIdx.x >> 5, lane = threadIdx.x & 31;
  const int b = blockIdx.y;
  const int s = blockIdx.x * 8 + wave;
  const bool active = (s < S);

  if (active) {
    int ci = perm[b * S + s];
    for (int c = lane; c < Cq; c += 32)
      csh[wave][c] = x[((long long)b * C + c) * P + ci];
  }
  __syncthreads();
  if (!active) return;

  float qs2 = 0.f;
  for (int c = lane; c < Cq; c += 32) { float v = csh[wave][c]; qs2 += v * v; }
#pragma unroll
  for (int o = 16; o > 0; o >>= 1) qs2 += __shfl_xor(qs2, o, 32);

  float bd[KK_]; int bi[KK_];
#pragma unroll
  for (int t = 0; t < KK_; ++t) { bd[t] = -3.4e38f; bi[t] = 0; }

  for (int p = lane; p < P; p += 32) {
    float inner = 0.f, q2 = 0.f;
    for (int c = 0; c < Cq; ++c) {
      float v = x[((long long)b * C + c) * P + p];
      inner += v * csh[wave][c];
      q2 += v * v;
    }
    float pd = 2.f * inner - q2 - qs2;
    if (pd > bd[KK_ - 1]) {
      float cd = pd; int cix = p;
#pragma unroll
      for (int t = 0; t < KK_; ++t) {
        if (cd > bd[t]) {
          float td = bd[t]; int ti = bi[t];
          bd[t] = cd; bi[t] = cix; cd = td; cix = ti;
        }
      }
    }
  }

#pragma unroll
  for (int off = 16; off > 0; off >>= 1) {
    float pb[KK_]; int pi_[KK_];
#pragma unroll
    for (int t = 0; t < KK_; ++t) {
      pb[t]  = __shfl_xor(bd[t], off, 32);
      pi_[t] = __shfl_xor(bi[t], off, 32);
    }
#pragma unroll
    for (int t = 0; t < KK_; ++t) {
      float cd = pb[t]; int cix = pi_[t];
      if (cd > bd[KK_ - 1]) {
#pragma unroll
        for (int u = 0; u < KK_; ++u) {
          if (cd > bd[u]) {
            float td = bd[u]; int ti = bi[u];
            bd[u] = cd; bi[u] = cix; cd = td; cix = ti;
          }
        }
      }
    }
  }

  if (lane == 0) {
#pragma unroll
    for (int t = 0; t < KK_; ++t)
      knn[((long long)b * S + s) * KK_ + t] = bi[t];
  }
}

// ---------------------------------------------------------------------------
// Gather with the reference's torch flatten/view scramble:
//   x1[b,c,s,k] = x[b,c, knn[b, (s*K+k)%S, (s*K+k)/S]]
// ---------------------------------------------------------------------------
__global__ void k_gather(const float* __restrict__ x, const int* __restrict__ knn,
                         float* __restrict__ x1, int B, int C, int P, int S, int K)
{
  long long i = (long long)blockIdx.x * blockDim.x + threadIdx.x;
  long long tot = (long long)B * C * S * K;
  if (i >= tot) return;
  int k = (int)(i % K);
  int s = (int)((i / K) % S);
  int c = (int)((i / ((long long)K * S)) % C);
  int b = (int)(i / ((long long)K * S * C));
  long long j = (long long)s * K + k;
  int sp = (int)(j % S), kp = (int)(j / S);
  int idx = knn[((long long)b * S + sp) * K + kp];
  x1[i] = x[((long long)b * C + c) * P + idx];
}

// PEmb: [PA(3), PC(3), diff(3), dist(1)] -> [B,10,S,K]
__global__ void k_pemb(const float* __restrict__ x1, float* __restrict__ pe,
                       int B, int C, int S, int K)
{
  int i = blockIdx.x * blockDim.x + threadIdx.x;
  if (i >= B * S * K) return;
  int k = i % K, s = (i / K) % S, b = i / (K * S);
  long long N = (long long)S * K;
  long long xb = (long long)b * C * N;
  float pa[3], pc[3], df[3];
  float d2 = 0.f;
#pragma unroll
  for (int c = 0; c < 3; ++c) {
    pa[c] = x1[xb + c * N + (long long)s * K + k];
    pc[c] = x1[xb + c * N + (long long)s * K + 0];
    df[c] = pa[c] - pc[c];
    d2 += df[c] * df[c];
  }
  long long ob = (long long)b * 10 * N + (long long)s * K + k;
#pragma unroll
  for (int c = 0; c < 3; ++c) {
    pe[ob + (0 + c) * N] = pa[c];
    pe[ob + (3 + c) * N] = pc[c];
    pe[ob + (6 + c) * N] = df[c];
  }
  pe[ob + 9 * N] = sqrtf(d2);
}

// GroupNorm + leaky over [B,C,N]; writes into out[B,Cout,N] at channel c0.
__global__ __launch_bounds__(256)
void k_gn_leaky(const float* __restrict__ in, float* __restrict__ out,
                const float* __restrict__ gamma, const float* __restrict__ beta,
                int C, int G, int N, int Cout, int c0, float slope)
{
  int bg = blockIdx.x;
  int g = bg % G, b = bg / G;
  int cpg = C / G;
  long long nelem = (long long)cpg * N;
  const float* ip = in + ((long long)b * C + (long long)g * cpg) * N;
  float s = 0.f, ss = 0.f;
  for (long long t = threadIdx.x; t < nelem; t += 256) {
    float v = ip[t]; s += v; ss += v * v;
  }
  __shared__ float sh1[256], sh2[256];
  sh1[threadIdx.x] = s; sh2[threadIdx.x] = ss;
  __syncthreads();
  for (int o = 128; o > 0; o >>= 1) {
    if (threadIdx.x < o) {
      sh1[threadIdx.x] += sh1[threadIdx.x + o];
      sh2[threadIdx.x] += sh2[threadIdx.x + o];
    }
    __syncthreads();
  }
  float mean = sh1[0] / (float)nelem;
  float var  = sh2[0] / (float)nelem - mean * mean;
  float r = rsqrtf(var + 1e-5f);
  float* op = out + ((long long)b * Cout + c0 + (long long)g * cpg) * N;
  for (long long t = threadIdx.x; t < nelem; t += 256) {
    int c = (int)(t / N);
    float v = (ip[t] - mean) * r * gamma[g * cpg + c] + beta[g * cpg + c];
    op[t] = (v >= 0.f) ? v : slope * v;
  }
}

// Row softmax in place (one block per row).
__global__ __launch_bounds__(256)
void k_softmax(float* __restrict__ X, int n)
{
  float* row = X + (long long)blockIdx.x * n;
  __shared__ float sh[256];
  int tid = threadIdx.x;
  float m = -3.4e38f;
  for (int t = tid; t < n; t += 256) m = fmaxf(m, row[t]);
  sh[tid] = m; __syncthreads();
  for (int o = 128; o > 0; o >>= 1) {
    if (tid < o) sh[tid] = fmaxf(sh[tid], sh[tid + o]);
    __syncthreads();
  }
  float M = sh[0]; __syncthreads();
  float s = 0.f;
  for (int t = tid; t < n; t += 256) {
    float e = __expf(row[t] - M); row[t] = e; s += e;
  }
  sh[tid] = s; __syncthreads();
  for (int o = 128; o > 0; o >>= 1) {
    if (tid < o) sh[tid] += sh[tid + o];
    __syncthreads();
  }
  float inv = 1.f / sh[0];
  for (int t = tid; t < n; t += 256) row[t] *= inv;
}

// Fused fe-attention: logits = w2 . leaky(att1,0.2), softmax over k,
// fsum[b,c,s] = sum_k feature[b,c,s,k]*att[k].
__global__ __launch_bounds__(128)
void k_fe_att(const float* __restrict__ feature, const float* __restrict__ att1,
              const float* __restrict__ w2, float* __restrict__ fsum,
              int B, int Cf, int S, int K)
{
  int bs = blockIdx.x;
  int s = bs % S, b = bs / S;
  __shared__ float lg[16];
  int tid = threadIdx.x;
  if (tid < K) {
    float acc = 0.f;
    for (int i2 = 0; i2 < 32; ++i2) {
      float v = att1[(((long long)b * 32 + i2) * S + s) * K + tid];
      v = (v >= 0.f) ? v : 0.2f * v;
      acc += w2[i2] * v;
    }
    lg[tid] = acc;
  }
  __syncthreads();
  float mx = -3.4e38f;
  for (int k = 0; k < K; ++k) mx = fmaxf(mx, lg[k]);
  float w[12]; float sm = 0.f;
  for (int k = 0; k < K; ++k) { w[k] = __expf(lg[k] - mx); sm += w[k]; }
  float inv = 1.f / sm;
  for (int c = tid; c < Cf; c += 128) {
    const float* fp = feature + (((long long)b * Cf + c) * S + s) * K;
    float acc = 0.f;
    for (int k = 0; k < K; ++k) acc += fp[k] * w[k];
    fsum[((long long)b * Cf + c) * S + s] = acc * inv;
  }
}

// out[b, oc0+c, s] = x1[b, c0+c, s, 0]
__global__ void k_slice(const float* __restrict__ x1, float* __restrict__ out,
                        int B, int Cin, int S, int K, int c0, int CC,
                        int Cout, int oc0)
{
  int i = blockIdx.x * blockDim.x + threadIdx.x;
  if (i >= B * CC * S) return;
  int s = i % S, c = (i / S) % CC, b = i / (S * CC);
  out[((long long)b * Cout + oc0 + c) * S + s] =
      x1[(((long long)b * Cin + c0 + c) * S + s) * (long long)K];
}

// out[b, oc0+c, s] = leaky(a+b2, slope)
__global__ void k_add_leaky_cat(const float* __restrict__ a,
                                const float* __restrict__ b2,
                                float* __restrict__ out,
                                int B, int C, int S, int Cout, int oc0,
                                float slope)
{
  int i = blockIdx.x * blockDim.x + threadIdx.x;
  if (i >= B * C * S) return;
  int s = i % S, c = (i / S) % C, b = i / (S * C);
  long long e = ((long long)b * C + c) * S + s;
  float v = a[e] + b2[e];
  out[((long long)b * Cout + oc0 + c) * S + s] = (v >= 0.f) ? v : slope * v;
}

// ===========================================================================
// Host side
// ===========================================================================
extern "C" void kernel_launch(void* const* d_in, const int* in_sizes, int n_in,
                              void* d_out, int out_size, void* d_ws, size_t ws_size,
                              hipStream_t stream)
{
  (void)in_sizes; (void)out_size; (void)ws_size;

  // Input flatten order (jax tree, sorted keys):
  //   0: x_t   1: x_s
  //   2..10 : cross  [bk,bq,bv,gk,gq,gv,wk,wq,wv]
  //   11..50: ext layers 0..3, each
  //           [b_feat,b_pos,g_feat,g_pos,w_att1,w_att2,w_b1,w_b2,w_feat,w_pos]
  //   51..59: self1  60..68: self2
  const float* F[70];
  for (int i = 0; i < 70 && i < n_in; ++i) F[i] = (const float*)d_in[i];
  const float* const* crossP = F + 2;
  const float* const* self1P = F + 51;
  const float* const* self2P = F + 60;

  // ---- workspace carve (fp32 slots, 256-aligned). Total ~73 MB. ----
  size_t off = 0;
  auto takef = [&](size_t n) -> float* {
    float* p = (float*)d_ws + off;
    off += (n + 255) & ~(size_t)255;
    return p;
  };
  float* xA    = takef(440000);          // ping xout (max 2*132*1638)
  float* xB    = takef(440000);          // pong xout
  float* ftBuf = takef(2 * 1028 * 256);  // L4 out, x_t pass
  float* fsBuf = takef(2 * 1028 * 256);  // L4 out, x_s pass
  float* x1buf = takef(2600064);         // max 2*132*9828
  float* pembB = takef(393216);          // 2*10*19656
  float* tmp1  = takef(2516992);         // conv pre-GN (pos)
  float* tmp2  = takef(2516992);         // conv pre-GN (feat)
  float* featB = takef(5033984);         // concat feature [B,outc,N]
  float* att1B = takef(1258496);         // 2*32*19656
  float* fsumB = takef(524288);
  float* x0B   = takef(262144);
  float* fea1B = takef(524288);
  float* fea2B = takef(524288);
  int*   permB = (int*)takef(4096);
  int*   knnB  = (int*)takef(39424);
  // transformer buffers alias extraction scratch (extraction is done by then)
  float* tq   = tmp1;
  float* tk   = tmp1 + 524288;
  float* tv   = tmp1 + 1048576;
  float* ttmp = tmp2;
  float* Hbuf = featB;                   // [2,8,256,256]
  float* y1   = featB + 1048576;         // [2,1024,256]
  float* y2   = featB + 1572864;         // [2,1024,256]

  auto gemm = [&](const float* A, const float* Bm, float* Cm,
                  int M, int N, int K, int lda, int ldb, int ldc,
                  long long sA, long long sB, long long sC,
                  int nb, int tA, int tB, float alpha) {
    dim3 g((N + GEMM_BN - 1) / GEMM_BN, (M + GEMM_BM - 1) / GEMM_BM, nb);
    k_gemm_f32<<<g, dim3(128), 0, stream>>>(A, Bm, Cm, M, N, K, lda, ldb, ldc,
                                            sA, sB, sC, tA, tB, alpha);
  };
  auto gn = [&](const float* in, float* out, const float* gamma,
                const float* beta, int C, int G, int N, int Cout, int c0,
                float slope) {
    k_gn_leaky<<<dim3(2 * G), dim3(256), 0, stream>>>(in, out, gamma, beta, C,
                                                      G, N, Cout, c0, slope);
  };

  // ---- feature extraction (4 layers) ----
  auto run_extract = [&](const float* x0, float* xout4, unsigned sb) {
    const int SN[4]   = {1638, 819, 409, 256};
    const int KA[4]   = {12, 12, 12, 4};
    const int GR[4]   = {4, 16, 16, 16};
    const int INCA[4] = {4, 128, 256, 512};
    const int OUTA[4] = {128, 256, 512, 1024};
    const float* x = x0;
    int Cin = 4, P = 16384;
    for (int L = 0; L < 4; ++L) {
      const int S = SN[L], K = KA[L], G = GR[L], inc = INCA[L], outc = OUTA[L];
      const int N = S * K;
      const float* const* par = F + 11 + L * 10;
      float* xout = (L == 3) ? xout4 : ((L & 1) ? xB : xA);
      const int Cq = (Cin == 4) ? 3 : Cin;

      { int tot = 2 * S;
        k_perm<<<dim3((tot + 255) / 256), dim3(256), 0, stream>>>(
            permB, 2, P, S, sb + (unsigned)L); }
      { dim3 g((S + 7) / 8, 2);
        if (K == 12)
          k_knn<12><<<g, dim3(256), 0, stream>>>(x, permB, knnB, 2, Cin, Cq, P, S);
        else
          k_knn<4><<<g, dim3(256), 0, stream>>>(x, permB, knnB, 2, Cin, Cq, P, S); }
      { long long tot = (long long)2 * Cin * S * K;
        k_gather<<<dim3((unsigned)((tot + 255) / 256)), dim3(256), 0, stream>>>(
            x, knnB, x1buf, 2, Cin, P, S, K); }
      { int tot = 2 * S * K;
        k_pemb<<<dim3((tot + 255) / 256), dim3(256), 0, stream>>>(
            x1buf, pembB, 2, Cin, S, K); }

      // pos branch: conv2(w_pos) -> GN+leaky(0.2) -> feature[:, outc/2:]
      gemm(par[9], pembB, tmp1, outc / 2, N, 10, 10, N, N,
           0, 10LL * N, (long long)(outc / 2) * N, 2, 0, 0, 1.f);
      gn(tmp1, featB, par[3], par[1], outc / 2, G, N, outc, outc / 2, 0.2f);

      // feat branch: conv2(w_feat) on x1[:,4:] (or x1 if Cin==4)
      const float* fin = (Cin == 4) ? (const float*)x1buf
                                    : (const float*)(x1buf + 4LL * N);
      gemm(par[8], fin, tmp2, outc / 2, N, inc, inc, N, N,
           0, (long long)Cin * N, (long long)(outc / 2) * N, 2, 0, 0, 1.f);
      gn(tmp2, featB, par[2], par[0], outc / 2, G, N, outc, 0, 0.2f);

      // attention pool over k
      gemm(par[4], featB, att1B, 32, N, outc, outc, N, N,
           0, (long long)outc * N, 32LL * N, 2, 0, 0, 1.f);
      k_fe_att<<<dim3(2 * S), dim3(128), 0, stream>>>(featB, att1B, par[5],
                                                      fsumB, 2, outc, S, K);

      // x0 = x1[:, 4:, :, 0] (or x1[:, :, :, 0] if Cin==4), compacted
      { int c0 = (Cin == 4) ? 0 : 4;
        int tot = 2 * inc * S;
        k_slice<<<dim3((tot + 255) / 256), dim3(256), 0, stream>>>(
            x1buf, x0B, 2, Cin, S, K, c0, inc, inc, 0); }

      gemm(par[6], fsumB, fea1B, outc, S, outc, outc, S, S,
           0, (long long)outc * S, (long long)outc * S, 2, 0, 0, 1.f);
      gemm(par[7], x0B, fea2B, outc, S, inc, inc, S, S,
           0, (long long)inc * S, (long long)outc * S, 2, 0, 0, 1.f);

      // concat [coords(4), leaky(Fea1+Fea2, 0.1)]
      { int tot = 2 * 4 * S;
        k_slice<<<dim3((tot + 255) / 256), dim3(256), 0, stream>>>(
            x1buf, xout, 2, Cin, S, K, 0, 4, 4 + outc, 0); }
      { int tot = 2 * outc * S;
        k_add_leaky_cat<<<dim3((tot + 255) / 256), dim3(256), 0, stream>>>(
            fea1B, fea2B, xout, 2, outc, S, 4 + outc, 4, 0.1f); }

      x = xout; Cin = 4 + outc; P = S;
    }
  };

  run_extract((const float*)d_in[0], ftBuf, 0u);   // ft from x_t
  run_extract((const float*)d_in[1], fsBuf, 16u);  // fs from x_s

  // ---- transformer: Q/K/V convs + GN + per-head Gram/softmax/out ----
  // tf indices (sorted): bk=0,bq=1,bv=2,gk=3,gq=4,gv=5,wk=6,wq=7,wv=8
  auto run_tf = [&](const float* const* tf, const float* Xq, long long sXq,
                    const float* Xk, long long sXk, float* out) {
    const float alpha_h = 1.0f / sqrtf(128.0f);
    // Q = leaky(GN(wq x Xq), 0.1)
    gemm(tf[7], Xq, ttmp, 1024, 256, 1024, 1024, 256, 256,
         0, sXq, 1024LL * 256, 2, 0, 0, 1.f);
    gn(ttmp, tq, tf[4], tf[1], 1024, 32, 256, 1024, 0, 0.1f);
    // K from Xk
    gemm(tf[6], Xk, ttmp, 1024, 256, 1024, 1024, 256, 256,
         0, sXk, 1024LL * 256, 2, 0, 0, 1.f);
    gn(ttmp, tk, tf[3], tf[0], 1024, 32, 256, 1024, 0, 0.1f);
    // V from Xq
    gemm(tf[8], Xq, ttmp, 1024, 256, 1024, 1024, 256, 256,
         0, sXq, 1024LL * 256, 2, 0, 0, 1.f);
    gn(ttmp, tv, tf[5], tf[2], 1024, 32, 256, 1024, 0, 0.1f);
    // H[z=b*8+h] = (Kh^T Vh)/sqrt(128): M=256,N=256,K=128, transA
    gemm(tk, tv, Hbuf, 256, 256, 128, 256, 256, 256,
         128LL * 256, 128LL * 256, 256LL * 256, 16, 1, 0, alpha_h);
    k_softmax<<<dim3(2 * 8 * 256), dim3(256), 0, stream>>>(Hbuf, 256);
    // out[z] = Qh x H^T: M=128,N=256,K=256, transB
    gemm(tq, Hbuf, out, 128, 256, 256, 256, 256, 256,
         128LL * 256, 256LL * 256, 128LL * 256, 16, 0, 1, 1.f);
  };

  // ft/fs feature views: channels 4.. of [B,1028,S=256]
  const float* ftp = ftBuf + 4LL * 256;
  const float* fsp = fsBuf + 4LL * 256;
  const long long sExt = 1028LL * 256;
  const long long sY   = 1024LL * 256;

  run_tf(crossP, fsp, sExt, ftp, sExt, y1);              // cross(fs, ft)
  run_tf(self1P, y1, sY, y1, sY, y2);                    // self1(y, y)
  run_tf(self2P, y2, sY, y2, sY, (float*)d_out);         // self2 -> d_out
}